// BottomUpAttention_20151986553271
// MI455X (gfx1250) — compile-verified
//
#include <hip/hip_runtime.h>

// BottomUpAttention (degenerate): softmax over a [n,1] axis makes weights==1,
// so out = tissue_features + segment_sum(cell_features, cluster_assignments).
// Memory-bound: ~512MB of cell rows streamed once -> ~23us floor @ 23.3 TB/s.
// Strategy: counting-sort cells by tissue, then per-tissue gather-sum with a
// software-pipelined GLOBAL_LOAD_ASYNC_TO_LDS_B128 ring (CDNA5 async path).

#define DIM        256
#define THREADS_WG 64      // 2 waves; each thread owns one float4 of a row
#define PIPE       8       // async ops in flight per wave
#define RING       16      // LDS ring slots (2x PIPE -> no WAR on slot reuse)

// ---- CDNA5 async-copy primitives (inline asm; portable across toolchains) ----
__device__ __forceinline__ void async_load_row16(unsigned lds_off, const void* gaddr) {
  // per-lane: copies 16 bytes from the global address gaddr to the LDS address lds_off; tracked by ASYNCcnt
  asm volatile("global_load_async_to_lds_b128 %0, %1, off"
               :: "v"(lds_off), "v"(gaddr) : "memory");
}

#if defined(__has_builtin) && __has_builtin(__builtin_amdgcn_s_wait_asynccnt)
#define S_WAIT_ASYNCCNT(N) do { __builtin_amdgcn_s_wait_asynccnt(N); \
                                asm volatile("" ::: "memory"); } while (0)
#else
#define S_WAIT_ASYNCCNT(N) asm volatile("s_wait_asynccnt %0" :: "n"(N) : "memory")
#endif

// ------------------------------ phase kernels ------------------------------
__global__ void bua_zero_counts(int* __restrict__ counts, int n_tissue) {
  int i = blockIdx.x * blockDim.x + threadIdx.x;
  if (i < n_tissue) counts[i] = 0;
}

__global__ void bua_hist(const int* __restrict__ assign, int* __restrict__ counts,
                         int n_cell) {
  int i = blockIdx.x * blockDim.x + threadIdx.x;
  if (i < n_cell) atomicAdd(&counts[assign[i]], 1);
}

// single block, 256 threads: exclusive scan of counts -> offsets[ n_tissue+1 ],
// and cursor[t] = offsets[t] for the scatter pass.
__global__ __launch_bounds__(256)
void bua_scan(const int* __restrict__ counts, int* __restrict__ offsets,
              int* __restrict__ cursor, int n_tissue) {
  __shared__ int partial[256];
  __shared__ int prefix[256];
  const int t = threadIdx.x;
  const int chunk = (n_tissue + 255) / 256;
  const int b0 = t * chunk;
  const int b1 = (b0 + chunk < n_tissue) ? (b0 + chunk) : n_tissue;
  int s = 0;
  for (int i = b0; i < b1; ++i) s += counts[i];
  partial[t] = s;
  __syncthreads();
  if (t == 0) {
    int run = 0;
    for (int i = 0; i < 256; ++i) { prefix[i] = run; run += partial[i]; }
    offsets[n_tissue] = run;          // == n_cell
  }
  __syncthreads();
  int off = prefix[t];
  for (int i = b0; i < b1; ++i) {
    offsets[i] = off;
    cursor[i]  = off;
    off += counts[i];
  }
}

__global__ void bua_scatter(const int* __restrict__ assign, int* __restrict__ cursor,
                            int* __restrict__ sorted, int n_cell) {
  int i = blockIdx.x * blockDim.x + threadIdx.x;
  if (i < n_cell) {
    int t = assign[i];
    int pos = atomicAdd(&cursor[t], 1);
    sorted[pos] = i;
  }
}

// One workgroup per tissue row. Each thread owns 16 bytes (float4) of the
// 1KB row. Cell rows are staged into an LDS ring via async b128 loads
// (each wave stages/consumes only its own lanes' data -> no barriers needed),
// accumulated in registers, then out = tissue_row + acc.
__global__ __launch_bounds__(THREADS_WG)
void bua_segsum(const float* __restrict__ cell, const float* __restrict__ tissue,
                const int* __restrict__ sorted, const int* __restrict__ offsets,
                float* __restrict__ out) {
  __shared__ float4 stage[RING * THREADS_WG];   // 16KB ring

  const int t   = blockIdx.x;
  const int tid = threadIdx.x;
  const int beg = offsets[t];
  const int n   = offsets[t + 1] - beg;

  const unsigned lds_lane =
      (unsigned)(unsigned long long)(&stage[0]) + (unsigned)tid * 16u;
  const char* cell_lane = (const char*)cell + (unsigned long long)tid * 16ull;

  // prologue: fill the pipe
  const int pre = (n < PIPE) ? n : PIPE;
  for (int i = 0; i < pre; ++i) {
    const int row = sorted[beg + i];
    async_load_row16(lds_lane + (unsigned)(i & (RING - 1)) * (DIM * 4u),
                     cell_lane + (unsigned long long)row * (DIM * 4ull));
  }

  float4 acc; acc.x = 0.f; acc.y = 0.f; acc.z = 0.f; acc.w = 0.f;

  // steady state: exactly PIPE ops in flight; wait(PIPE-1) => oldest complete
  const int nMain = (n > PIPE) ? (n - PIPE) : 0;
  for (int i = 0; i < nMain; ++i) {
    S_WAIT_ASYNCCNT(PIPE - 1);
    const float4 v = stage[(i & (RING - 1)) * THREADS_WG + tid];
    acc.x += v.x; acc.y += v.y; acc.z += v.z; acc.w += v.w;
    const int row = sorted[beg + i + PIPE];
    async_load_row16(lds_lane + (unsigned)((i + PIPE) & (RING - 1)) * (DIM * 4u),
                     cell_lane + (unsigned long long)row * (DIM * 4ull));
  }

  // drain
  S_WAIT_ASYNCCNT(0);
  for (int i = nMain; i < n; ++i) {
    const float4 v = stage[(i & (RING - 1)) * THREADS_WG + tid];
    acc.x += v.x; acc.y += v.y; acc.z += v.z; acc.w += v.w;
  }

  const float4 tv = ((const float4*)tissue)[t * (DIM / 4) + tid];
  float4 o;
  o.x = tv.x + acc.x; o.y = tv.y + acc.y; o.z = tv.z + acc.z; o.w = tv.w + acc.w;
  ((float4*)out)[t * (DIM / 4) + tid] = o;
}

// ------------------------------ host wrapper ------------------------------
extern "C" void kernel_launch(void* const* d_in, const int* in_sizes, int n_in,
                              void* d_out, int out_size, void* d_ws, size_t ws_size,
                              hipStream_t stream) {
  const float* cell   = (const float*)d_in[0];   // [n_cell, 256]
  const float* tissue = (const float*)d_in[1];   // [n_tissue, 256]
  const int*   assign = (const int*)d_in[2];     // [n_cell]
  // d_in[3..7] (W_cell, b_cell, W_tissue, b_tissue, attn_w) are dead code:
  // softmax over a singleton axis makes the attention weights identically 1.
  float* outp = (float*)d_out;

  const int n_cell   = in_sizes[0] / DIM;
  const int n_tissue = in_sizes[1] / DIM;

  // workspace layout (ints): counts | offsets(+1) | cursor | sorted
  int* ws      = (int*)d_ws;
  int* counts  = ws;
  int* offsets = counts + n_tissue;
  int* cursor  = offsets + n_tissue + 1;
  int* sorted  = cursor + n_tissue;
  (void)ws_size; (void)n_in; (void)out_size;

  const int tb = 256;
  bua_zero_counts<<<(n_tissue + tb - 1) / tb, tb, 0, stream>>>(counts, n_tissue);
  bua_hist<<<(n_cell + tb - 1) / tb, tb, 0, stream>>>(assign, counts, n_cell);
  bua_scan<<<1, 256, 0, stream>>>(counts, offsets, cursor, n_tissue);
  bua_scatter<<<(n_cell + tb - 1) / tb, tb, 0, stream>>>(assign, cursor, sorted, n_cell);
  bua_segsum<<<n_tissue, THREADS_WG, 0, stream>>>(cell, tissue, sorted, offsets, outp);
}